// TranslationDown_10024453668973
// MI455X (gfx1250) — compile-verified
//
#include <hip/hip_runtime.h>
#include <cfloat>

#define B_   4
#define N_   8192
#define C_   64
#define K_   16
#define OUT_ 128
#define S_   4096   // N/2

typedef __attribute__((ext_vector_type(2))) float v2f;
typedef __attribute__((ext_vector_type(8))) float v8f;

// ---------------------------------------------------------------------------
// Kernel 1: Furthest point sampling. One workgroup (1024 threads) per batch.
// Each thread owns 8 points + their running min-distance entirely in registers.
// Argmax per step: wave32 shfl reduction (no barrier) + 32-entry LDS reduction
// => 2 barriers per serial step. Tie-break: lowest index (matches jnp.argmax).
// ---------------------------------------------------------------------------
__global__ __launch_bounds__(1024) void fps_kernel(const float* __restrict__ pc,
                                                   int* __restrict__ fps_idx) {
  const int b = blockIdx.x;
  const int t = threadIdx.x;
  const float* px = pc + (size_t)b * 3 * N_;
  const float* py = px + N_;
  const float* pz = py + N_;

  float x[8], y[8], z[8], d[8];
#pragma unroll
  for (int j = 0; j < 8; ++j) {
    const int n = t + j * 1024;           // strided -> coalesced loads
    x[j] = px[n]; y[j] = py[n]; z[j] = pz[n];
    d[j] = FLT_MAX;
  }

  __shared__ float rv[32];
  __shared__ int   ri[32];
  __shared__ float sl[3];

  if (t == 0) {
    fps_idx[b * S_ + 0] = 0;
    sl[0] = px[0]; sl[1] = py[0]; sl[2] = pz[0];
  }
  __syncthreads();

  const int lane = t & 31;
  const int wid  = t >> 5;

  for (int step = 1; step < S_; ++step) {
    const float bx = sl[0], by = sl[1], bz = sl[2];
    float bestv = -1.0f; int besti = 0;
#pragma unroll
    for (int j = 0; j < 8; ++j) {
      const float dx = x[j] - bx, dy = y[j] - by, dz = z[j] - bz;
      const float dd = dx * dx + dy * dy + dz * dz;
      d[j] = fminf(d[j], dd);
      if (d[j] > bestv) { bestv = d[j]; besti = t + j * 1024; }
    }
    // wave32 (v, idx) argmax reduction, prefer lower index on ties
#pragma unroll
    for (int off = 16; off > 0; off >>= 1) {
      const float v2 = __shfl_xor(bestv, off, 32);
      const int   i2 = __shfl_xor(besti, off, 32);
      if (v2 > bestv || (v2 == bestv && i2 < besti)) { bestv = v2; besti = i2; }
    }
    if (lane == 0) { rv[wid] = bestv; ri[wid] = besti; }
    __syncthreads();
    if (wid == 0) {
      bestv = rv[lane]; besti = ri[lane];
#pragma unroll
      for (int off = 16; off > 0; off >>= 1) {
        const float v2 = __shfl_xor(bestv, off, 32);
        const int   i2 = __shfl_xor(besti, off, 32);
        if (v2 > bestv || (v2 == bestv && i2 < besti)) { bestv = v2; besti = i2; }
      }
      if (lane == 0) {
        fps_idx[b * S_ + step] = besti;
        sl[0] = px[besti]; sl[1] = py[besti]; sl[2] = pz[besti];
      }
    }
    __syncthreads();
  }
}

// ---------------------------------------------------------------------------
// Kernel 2: write fps_pts [B,3,S] (floats) and fps_idx (int64) outputs.
// ---------------------------------------------------------------------------
__global__ __launch_bounds__(256) void fps_out_kernel(const float* __restrict__ pc,
                                                      const int* __restrict__ fps_idx,
                                                      float* __restrict__ out_pts,
                                                      long long* __restrict__ out_idx) {
  const int q = blockIdx.x * blockDim.x + threadIdx.x;
  if (q >= B_ * S_) return;
  const int b = q >> 12;            // / S_
  const int s = q & (S_ - 1);
  const int idx = fps_idx[q];
  const float* base = pc + (size_t)b * 3 * N_;
  out_pts[(size_t)b * 3 * S_ + 0 * S_ + s] = base[idx];
  out_pts[(size_t)b * 3 * S_ + 1 * S_ + s] = base[N_ + idx];
  out_pts[(size_t)b * 3 * S_ + 2 * S_ + s] = base[2 * N_ + idx];
  out_idx[q] = (long long)idx;
}

// ---------------------------------------------------------------------------
// Kernel 3: KNN (K=16 smallest squared distances, self included).
// 256 queries per block; candidate points tiled through LDS (12KB/tile).
// Register insertion list, strict '<' keeps lower index on ties (top_k stable).
// ---------------------------------------------------------------------------
#define KTILE 1024
__global__ __launch_bounds__(256) void knn_kernel(const float* __restrict__ pc,
                                                  const int* __restrict__ fps_idx,
                                                  int* __restrict__ knn) {
  const int tid = threadIdx.x;
  const int q = blockIdx.x * 256 + tid;
  const int b = q >> 12;
  const float* px = pc + (size_t)b * 3 * N_;
  const float* py = px + N_;
  const float* pz = py + N_;
  const int qi = fps_idx[q];
  const float qx = px[qi], qy = py[qi], qz = pz[qi];

  float bd[K_]; int bi[K_];
#pragma unroll
  for (int j = 0; j < K_; ++j) { bd[j] = FLT_MAX; bi[j] = 0; }

  __shared__ float sx[KTILE], sy[KTILE], sz[KTILE];

  for (int t0 = 0; t0 < N_; t0 += KTILE) {
    __syncthreads();
#pragma unroll
    for (int j = 0; j < KTILE / 256; ++j) {
      const int i = tid + j * 256;
      sx[i] = px[t0 + i]; sy[i] = py[t0 + i]; sz[i] = pz[t0 + i];
    }
    __syncthreads();
    for (int i = 0; i < KTILE; ++i) {
      const float dx = sx[i] - qx, dy = sy[i] - qy, dz = sz[i] - qz;
      const float d2 = dx * dx + dy * dy + dz * dz;
      if (d2 < bd[K_ - 1]) {
        float cd = d2; int ci = t0 + i;
#pragma unroll
        for (int j = 0; j < K_; ++j) {
          if (cd < bd[j]) {
            const float td = bd[j]; const int ti = bi[j];
            bd[j] = cd; bi[j] = ci; cd = td; ci = ti;
          }
        }
      }
    }
  }
#pragma unroll
  for (int j = 0; j < K_; ++j) knn[(size_t)q * K_ + j] = bi[j];
}

// ---------------------------------------------------------------------------
// Kernel 4: grouped 1x1 conv + ReLU + max over K via fp32 WMMA.
// One wave per (b,s). A tile = 16 neighbors x 64 channels, preloaded into
// registers in the native V_WMMA_F32_16X16X4_F32 A layout:
//   lanes 0-15 hold (K=0,K=1) of row M=lane; lanes 16-31 hold (K=2,K=3).
// B tile read as contiguous float2 pairs from conv_w[OUT][C].
// 8 output tiles of 16 columns; max over M = max over 8 acc VGPRs + shfl_xor(16).
// ---------------------------------------------------------------------------
__global__ __launch_bounds__(256) void group_gemm_kernel(const float* __restrict__ feat,
                                                         const float* __restrict__ w,
                                                         const float* __restrict__ bias,
                                                         const int* __restrict__ knn,
                                                         float* __restrict__ y) {
  const int lane = threadIdx.x & 31;
  const int wave = (blockIdx.x * blockDim.x + threadIdx.x) >> 5;
  const int b = wave >> 12;            // / S_
  const int s = wave & (S_ - 1);
  const int half = lane >> 4;          // 0 -> K=0,1 ; 1 -> K=2,3
  const int n16 = lane & 15;

  const int idx = knn[((size_t)b * S_ + s) * K_ + n16];   // A-row = neighbor n16
  const float* fb = feat + (size_t)b * C_ * N_ + idx;

  v2f a[16];
#pragma unroll
  for (int t = 0; t < 16; ++t) {
    const int c0 = 4 * t + half * 2;
    a[t][0] = fb[(size_t)c0 * N_];
    a[t][1] = fb[(size_t)(c0 + 1) * N_];
  }

  const v2f* w2 = (const v2f*)w;       // conv_w row-major: (o, 2c)/(o, 2c+1) pairs
#pragma unroll
  for (int ot = 0; ot < OUT_ / 16; ++ot) {
    v8f acc = {};
#pragma unroll
    for (int t = 0; t < 16; ++t) {
      const int o = ot * 16 + n16;     // B column = output channel
      const v2f bv = w2[o * (C_ / 2) + 2 * t + half];
      acc = __builtin_amdgcn_wmma_f32_16x16x4_f32(
          /*neg_a=*/false, a[t], /*neg_b=*/false, bv,
          /*c_mod=*/(short)0, acc, /*reuse_a=*/false, /*reuse_b=*/false);
    }
    // max over K (= matrix M dim): 8 VGPRs hold M=0..7 (lanes<16) / M=8..15
    float m = acc[0];
#pragma unroll
    for (int i = 1; i < 8; ++i) m = fmaxf(m, acc[i]);
    m = fmaxf(m, __shfl_xor(m, 16, 32));
    const float val = fmaxf(m + bias[ot * 16 + n16], 0.0f);
    if (lane < 16) {
      y[(size_t)b * OUT_ * S_ + (size_t)(ot * 16 + n16) * S_ + s] = val;
    }
  }
}

// ---------------------------------------------------------------------------
extern "C" void kernel_launch(void* const* d_in, const int* in_sizes, int n_in,
                              void* d_out, int out_size, void* d_ws, size_t ws_size,
                              hipStream_t stream) {
  const float* pc      = (const float*)d_in[0];  // [B,3,N]
  const float* feature = (const float*)d_in[1];  // [B,C,N]
  const float* conv_w  = (const float*)d_in[2];  // [OUT,C]
  const float* conv_b  = (const float*)d_in[3];  // [OUT]

  float* outf     = (float*)d_out;
  float* out_pts  = outf;                                   // B*3*S floats
  float* out_y    = outf + (size_t)B_ * 3 * S_;             // B*OUT*S floats
  long long* out_idx =
      (long long*)(outf + (size_t)B_ * 3 * S_ + (size_t)B_ * OUT_ * S_);  // B*S i64

  int* fps_idx = (int*)d_ws;                                // B*S ints
  int* knn = (int*)((char*)d_ws + (((size_t)B_ * S_ * sizeof(int) + 255) & ~(size_t)255));

  hipLaunchKernelGGL(fps_kernel, dim3(B_), dim3(1024), 0, stream, pc, fps_idx);
  hipLaunchKernelGGL(fps_out_kernel, dim3(B_ * S_ / 256), dim3(256), 0, stream,
                     pc, fps_idx, out_pts, out_idx);
  hipLaunchKernelGGL(knn_kernel, dim3(B_ * S_ / 256), dim3(256), 0, stream,
                     pc, fps_idx, knn);
  hipLaunchKernelGGL(group_gemm_kernel, dim3(B_ * S_ * 32 / 256), dim3(256), 0, stream,
                     feature, conv_w, conv_b, knn, out_y);
}